// RNNWWD_80272938762402
// MI455X (gfx1250) — compile-verified
//
#include <hip/hip_runtime.h>
#include <hip/hip_bf16.h>
#include <math.h>

// ---------------- problem constants ----------------
#define NODE    1024
#define BATCH   250
#define HIDDEN  15

// ---------------- decomposition ----------------
#define NBLK         64            // workgroups in main kernel (A rows split)
#define ROWS_PER_BLK 16            // NODE / NBLK
#define TPB          256           // 8 wave32 per block
#define WAVES        8
#define KCHUNK       128           // NODE / WAVES  (K slice per wave)
#define NTILE        32            // KCHUNK / 4    (WMMA k=4 tiles per wave)

typedef float v2f __attribute__((ext_vector_type(2)));
typedef float v8f __attribute__((ext_vector_type(8)));

// ---------------- workspace layout (float offsets) ----------------
constexpr size_t OFF_WSS   = 0;                          // w_s matrix  (1024*1024)
constexpr size_t OFF_ROW   = (size_t)NODE * NODE;        // rowsum(w_s) (1024)
constexpr size_t OFF_PARTS = OFF_ROW + NODE;             // 4096 block partial sums of squares
constexpr size_t OFF_SUMSQ = OFF_PARTS + 4096;           // 1 float
constexpr size_t OFF_EBUF  = OFF_SUMSQ + 4;              // double-buffered E (2*1024), 16B aligned
constexpr size_t OFF_BAR   = OFF_EBUF + 2 * NODE;        // 2 uints: {count, generation}

// ---------------- helpers ----------------
__device__ __forceinline__ float htf(float a, float b, float d, float z) {
    float lin = fmaf(a, z, -b);
    float num = 1e-5f + fabsf(lin);
    float den = 1e-5f * d + fabsf(1.0f - expf(-d * lin));
    return num / den;
}

// device-wide sense-reversing barrier; all threads of all NBLK blocks call it.
__device__ __forceinline__ unsigned gbar(unsigned* bar, unsigned lg) {
    unsigned target = lg + 1u;
    __threadfence();            // publish this thread's global stores device-wide
    __syncthreads();
    if (threadIdx.x == 0) {
        unsigned prev = __hip_atomic_fetch_add(&bar[0], 1u, __ATOMIC_ACQ_REL,
                                               __HIP_MEMORY_SCOPE_AGENT);
        if (prev == NBLK - 1) {
            __hip_atomic_store(&bar[0], 0u, __ATOMIC_RELAXED, __HIP_MEMORY_SCOPE_AGENT);
            __hip_atomic_store(&bar[1], target, __ATOMIC_RELEASE, __HIP_MEMORY_SCOPE_AGENT);
        } else {
            while (__hip_atomic_load(&bar[1], __ATOMIC_ACQUIRE,
                                     __HIP_MEMORY_SCOPE_AGENT) != target) {
                __builtin_amdgcn_s_sleep(2);
            }
        }
    }
    __syncthreads();
    return target;
}

// ---------------- precompute kernels ----------------
__global__ void k_init(unsigned* bar) {
    bar[0] = 0u;
    bar[1] = 0u;
}

// w_s[i][j] = log1p(0.5*(e^wbb[i,j]*sc[i,j] + e^wbb[j,i]*sc[j,i])); per-block sum of squares.
__global__ __launch_bounds__(256) void k_ws(const float* __restrict__ wbb,
                                            const float* __restrict__ sc,
                                            float* __restrict__ ws_s,
                                            float* __restrict__ parts) {
    __shared__ float red[256];
    int idx = blockIdx.x * 256 + threadIdx.x;   // grid = 4096 blocks -> 1M elems
    int i = idx >> 10, j = idx & 1023;
    float wij = expf(wbb[i * NODE + j]) * sc[i * NODE + j];
    float wji = expf(wbb[j * NODE + i]) * sc[j * NODE + i];
    float v = log1pf(0.5f * (wij + wji));
    ws_s[idx] = v;
    red[threadIdx.x] = v * v;
    __syncthreads();
    for (int st = 128; st > 0; st >>= 1) {
        if (threadIdx.x < st) red[threadIdx.x] += red[threadIdx.x + st];
        __syncthreads();
    }
    if (threadIdx.x == 0) parts[blockIdx.x] = red[0];
}

// deterministic second-pass reduction of the 4096 partials.
__global__ __launch_bounds__(256) void k_sumsq(const float* __restrict__ parts,
                                               float* __restrict__ sumsq) {
    __shared__ float red[256];
    float s = 0.f;
    for (int i = 0; i < 16; ++i) s += parts[threadIdx.x + 256 * i];
    red[threadIdx.x] = s;
    __syncthreads();
    for (int st = 128; st > 0; st >>= 1) {
        if (threadIdx.x < st) red[threadIdx.x] += red[threadIdx.x + st];
        __syncthreads();
    }
    if (threadIdx.x == 0) sumsq[0] = red[0];
}

__global__ __launch_bounds__(256) void k_rowsum(const float* __restrict__ ws_s,
                                                float* __restrict__ rowsum) {
    __shared__ float red[256];
    int row = blockIdx.x;
    float s = 0.f;
    for (int k = 0; k < 4; ++k) s += ws_s[row * NODE + threadIdx.x + 256 * k];
    red[threadIdx.x] = s;
    __syncthreads();
    for (int st = 128; st > 0; st >>= 1) {
        if (threadIdx.x < st) red[threadIdx.x] += red[threadIdx.x + st];
        __syncthreads();
    }
    if (threadIdx.x == 0) rowsum[row] = red[0];
}

// ---------------- main persistent simulation kernel ----------------
__global__ __launch_bounds__(TPB) void wwd_main(const float* __restrict__ ext,    // (N,H,B)
                                                const float* __restrict__ inp,    // (N,H,B,2)
                                                const float* __restrict__ nout,   // (N,B)
                                                const float* __restrict__ hx,     // (N,6)
                                                const float* __restrict__ ws_s,   // (N,N)
                                                const float* __restrict__ rowsum, // (N)
                                                const float* __restrict__ sumsq,  // scalar
                                                float* __restrict__ ebuf,         // (2,N)
                                                unsigned* __restrict__ bar,       // {cnt,gen}
                                                float* __restrict__ out) {        // (B,N)
    __shared__ __align__(16) float E_lds[NODE];
    __shared__ float part[WAVES * 16];

    const int tid  = threadIdx.x;
    const int blk  = blockIdx.x;
    const int wave = tid >> 5;
    const int lane = tid & 31;
    const int hs   = (lane >> 4) & 1;          // 0: lanes 0-15, 1: lanes 16-31
    const int row  = blk * ROWS_PER_BLK + (lane & 15);   // A-matrix row of this lane

    const float inv_norm = 1.0f / sqrtf(sumsq[0]);

    // --- preload this wave's 16x128 A-slice into VGPRs in WMMA A-operand layout ---
    // A[m][k] = (w_s[m][k] - (m==k)*rowsum[m]) * inv_norm
    v2f areg[NTILE];
    {
        const float rs = rowsum[row];
#pragma unroll
        for (int t = 0; t < NTILE; ++t) {
            int k = wave * KCHUNK + t * 4 + hs * 2;
            float ax = ws_s[row * NODE + k];
            float ay = ws_s[row * NODE + k + 1];
            if (k == row)     ax -= rs;
            if (k + 1 == row) ay -= rs;
            areg[t].x = ax * inv_norm;
            areg[t].y = ay * inv_norm;
        }
    }

    // --- per-node state (threads 0..15 each own one node) ---
    const int node = blk * ROWS_PER_BLK + tid;   // only meaningful for tid<16
    float I_s = 0.f, x_s = 0.f, f_s = 1.f, v_s = 1.f, q_s = 1.f;
    if (tid < ROWS_PER_BLK) {
        const float* h = hx + node * 6;
        float E0 = h[0];
        I_s = h[1]; x_s = h[2]; f_s = h[3]; v_s = h[4]; q_s = h[5];
        ebuf[node] = E0;                         // E buffer parity 0
    }

    unsigned lg = 0;
    lg = gbar(bar, lg);                          // init E visible everywhere

    const float sqrt_dt = 0.22360679775f;        // sqrt(0.05)
    int s = 0;                                   // global step index
    for (int b = 0; b < BATCH; ++b) {
        // hoist the per-TR BOLD noise load off the critical path
        float nb = 0.f;
        if (tid < ROWS_PER_BLK) nb = nout[node * BATCH + b];

        for (int h = 0; h < HIDDEN; ++h) {
            // ---- prefetch noise/drive for this step: independent of E, so the
            //      global_loads issue here and complete under the WMMA chain ----
            float nE = 0.f, nI = 0.f, u = 0.f;
            if (tid < ROWS_PER_BLK) {
                const int base = (node * HIDDEN + h) * BATCH + b;
                nE = inp[2 * base];
                nI = inp[2 * base + 1];
                u  = ext[base];
            }

            // broadcast E into LDS (coalesced float4)
            const float4 ev = ((const float4*)(ebuf + (s & 1) * NODE))[tid];
            ((float4*)E_lds)[tid] = ev;
            __syncthreads();

            // y = A @ E via chained f32 WMMAs (B = E chunk replicated over all 16 cols)
            v8f acc = {0.f, 0.f, 0.f, 0.f, 0.f, 0.f, 0.f, 0.f};
#pragma unroll
            for (int t = 0; t < NTILE; ++t) {
                int kb = wave * KCHUNK + t * 4 + hs * 2;
                v2f bv;
                bv.x = E_lds[kb];
                bv.y = E_lds[kb + 1];
                acc = __builtin_amdgcn_wmma_f32_16x16x4_f32(
                    false, areg[t], false, bv, (short)0, acc, false, false);
            }
            // column N=0 of D lives in lane 0 (M=r) and lane 16 (M=8+r)
            if (lane == 0) {
#pragma unroll
                for (int r = 0; r < 8; ++r) part[wave * 16 + r] = acc[r];
            } else if (lane == 16) {
#pragma unroll
                for (int r = 0; r < 8; ++r) part[wave * 16 + 8 + r] = acc[r];
            }
            __syncthreads();

            if (tid < ROWS_PER_BLK) {
                float y = 0.f;
#pragma unroll
                for (int w = 0; w < WAVES; ++w) y += part[w * 16 + tid];

                const float E = E_lds[node];

                // g_EE = g_IE = g_EI = 0.001 + relu(0.1) = 0.101 ; W_E*I0=0.32 ; W_I*I0=0.224
                float IE = fmaxf(0.f, 0.32f + 0.101f * E + 20.0f * y - 0.101f * I_s) + u;
                float II = fmaxf(0.f, 0.224f + 0.101f * E - I_s);
                float rE = htf(310.f, 125.f, 0.16f,  IE);
                float rI = htf(615.f, 177.f, 0.087f, II);

                float ddE = E + 0.05f * (-E * 0.01f + 0.000641f * (1.f - E) * rE)
                              + sqrt_dt * nE * 0.02f;
                float ddI = I_s + 0.05f * (-I_s * 0.1f + 0.001f * rI)
                              + sqrt_dt * nI * 0.02f;
                float dx = x_s + 0.05f * (E - x_s * (1.f / 0.65f) - (f_s - 1.f) * (1.f / 0.41f));
                float df = f_s + 0.05f * x_s;
                // v^(1/alpha) = exp(3.125*ln v), v in (0,2); 0.66^(1/f) = exp(ln(0.66)/f)
                float vp = expf(3.125f * logf(v_s));
                float dv = v_s + 0.05f * (f_s - vp) * (1.f / 0.98f);
                float dq = q_s + 0.05f * (f_s * (1.f - expf(-0.4155154439616658f / f_s))
                                              * (1.f / 0.34f)
                                          - q_s * vp / v_s) * (1.f / 0.98f);

                float Enew = tanhf(1e-5f + fmaxf(0.f, ddE));
                I_s = tanhf(1e-5f + fmaxf(0.f, ddI));
                x_s = dx;
                f_s = 1.f + tanhf(df - 1.f);
                v_s = 1.f + tanhf(dv - 1.f);
                q_s = 1.f + tanhf(dq - 1.f);

                ebuf[((s + 1) & 1) * NODE + node] = Enew;
            }
            ++s;
            lg = gbar(bar, lg);                  // one device barrier per step
        }
        // BOLD readout for this TR
        if (tid < ROWS_PER_BLK) {
            float bold = 0.02f * nb
                       + (100.f * 0.02f / 0.34f)
                         * (2.38f * (1.f - q_s) + 2.0f * (1.f - q_s / v_s) + 0.48f * (1.f - v_s));
            out[b * NODE + node] = bold;
        }
    }
}

// ---------------- host-side launch ----------------
extern "C" void kernel_launch(void* const* d_in, const int* in_sizes, int n_in,
                              void* d_out, int out_size, void* d_ws, size_t ws_size,
                              hipStream_t stream) {
    (void)in_sizes; (void)n_in; (void)out_size; (void)ws_size;
    const float* ext  = (const float*)d_in[0];   // external (N,H,B)
    const float* inp  = (const float*)d_in[1];   // inp      (N,H,B,2)
    const float* nout = (const float*)d_in[2];   // noise_out(N,B)
    const float* hx   = (const float*)d_in[3];   // hx       (N,6)
    /* d_in[4] = hE   : dead state (delays deleted in reference) */
    const float* wbb  = (const float*)d_in[5];   // w_bb     (N,N)
    const float* sc   = (const float*)d_in[6];   // sc       (N,N)
    /* d_in[7] = dist : unused */

    float* ws       = (float*)d_ws;
    float* ws_s     = ws + OFF_WSS;
    float* rowsum   = ws + OFF_ROW;
    float* parts    = ws + OFF_PARTS;
    float* sumsq    = ws + OFF_SUMSQ;
    float* ebuf     = ws + OFF_EBUF;
    unsigned* barp  = (unsigned*)(ws + OFF_BAR);

    k_init  <<<dim3(1),    dim3(1),   0, stream>>>(barp);
    k_ws    <<<dim3(4096), dim3(256), 0, stream>>>(wbb, sc, ws_s, parts);
    k_sumsq <<<dim3(1),    dim3(256), 0, stream>>>(parts, sumsq);
    k_rowsum<<<dim3(1024), dim3(256), 0, stream>>>(ws_s, rowsum);
    wwd_main<<<dim3(NBLK), dim3(TPB), 0, stream>>>(ext, inp, nout, hx, ws_s, rowsum,
                                                   sumsq, ebuf, barp, (float*)d_out);
}